// MeshUnPool_memory_86474871537965
// MI455X (gfx1250) — compile-verified
//
#include <hip/hip_runtime.h>

// MeshUnPool feature restore: out[row, :] = features[parent_idx[row], :]
// Pure gather, bandwidth-bound (~2 GB traffic -> ~87us floor @ 23.3 TB/s).
//
// Wave32 layout: one wave per 256-float row (64 x float4). Lane l moves
// float4 slots l and l+32 -> two coalesced 512B bursts per load/store.
// Stores are non-temporal (1 GB write-once stream; keep L2 for the
// 512 MB features table which sees ~2x reuse). Next iteration's gathered
// row is prefetched (global_prefetch_b8) to hide random-access latency.

typedef __attribute__((ext_vector_type(4))) float v4f;

namespace {
constexpr int kChannels     = 256;
constexpr int kV4PerRow     = kChannels / 4;  // 64 float4 per row
constexpr int kWavesPerBlk  = 8;              // 256 threads = 8 wave32
constexpr int kBlockThreads = kWavesPerBlk * 32;
constexpr int kMaxBlocks    = 8192;           // 65536 waves in flight
}  // namespace

__global__ __launch_bounds__(kBlockThreads)
void meshunpool_gather_kernel(const float* __restrict__ feat,
                              const int* __restrict__ parent_idx,
                              float* __restrict__ out,
                              int n_rows) {
  const int lane = threadIdx.x & 31;
  const int wave = threadIdx.x >> 5;
  const int wave_stride = gridDim.x * kWavesPerBlk;

  int row = blockIdx.x * kWavesPerBlk + wave;

  for (; row < n_rows; row += wave_stride) {
    // Wave-uniform parent index (all lanes hit the same cacheline).
    const long long parent = (long long)parent_idx[row];
    const v4f* __restrict__ src = (const v4f*)feat + parent * kV4PerRow;

    // Prefetch next iteration's gathered row: each lane touches one byte at
    // its 32B-strided slot -> the whole 1024B row is pulled toward the caches
    // (global_prefetch_b8; speculative, cheap, no counter traffic).
    const int next_row = row + wave_stride;
    if (next_row < n_rows) {
      const long long next_parent = (long long)parent_idx[next_row];
      const char* p =
          (const char*)((const v4f*)feat + next_parent * kV4PerRow) + lane * 32;
      __builtin_prefetch(p, 0, 3);
    }

    // Two coalesced 128-bit loads per lane: slots [lane] and [lane+32]
    // cover float4 0..31 then 32..63 of the row.
    const v4f a = src[lane];
    const v4f b = src[lane + 32];

    v4f* __restrict__ dst = (v4f*)out + (long long)row * kV4PerRow;
    // Non-temporal 128-bit stores: write-once stream, don't pollute L2.
    __builtin_nontemporal_store(a, dst + lane);
    __builtin_nontemporal_store(b, dst + lane + 32);
  }
}

extern "C" void kernel_launch(void* const* d_in, const int* in_sizes, int n_in,
                              void* d_out, int out_size, void* d_ws, size_t ws_size,
                              hipStream_t stream) {
  const float* feat      = (const float*)d_in[0];  // [N_POOLED, 256] fp32
  const int* parent_idx  = (const int*)d_in[1];    // [N_UNPOOLED] int
  float* out             = (float*)d_out;          // [N_UNPOOLED, 256] fp32

  const int n_rows = in_sizes[1];

  int blocks = (n_rows + kWavesPerBlk - 1) / kWavesPerBlk;
  if (blocks > kMaxBlocks) blocks = kMaxBlocks;
  if (blocks < 1) blocks = 1;

  meshunpool_gather_kernel<<<blocks, kBlockThreads, 0, stream>>>(
      feat, parent_idx, out, n_rows);
}